// IncMaxPoolModule_74028056314078
// MI455X (gfx1250) — compile-verified
//
#include <hip/hip_runtime.h>
#include <stdint.h>

// Problem constants (static shapes from the reference setup).
#define BB      32
#define CC      64
#define HH      224
#define WW      224
#define HO      112
#define WO      112
#define OUT_PH  17      // ceil((32+2-1)/2)
#define OUT_PW  17
#define EXT_H   34      // (17-1)*2 + 2
#define EXT_W   34
#define N_OUT   (BB*CC*HO*WO)   // 25,690,112 floats

typedef __attribute__((ext_vector_type(4))) float f4;

// -----------------------------------------------------------------------------
// Kernel 1: bulk copy of the stale pooled output into d_out.
// Pure stream: 128-bit accesses, non-temporal both ways (no reuse).
// -----------------------------------------------------------------------------
__global__ void inc_maxpool_copy_kernel(const f4* __restrict__ src,
                                        f4* __restrict__ dst, int n4) {
    int i = blockIdx.x * blockDim.x + threadIdx.x;
    if (i < n4) {
        f4 v = __builtin_nontemporal_load(&src[i]);
        __builtin_nontemporal_store(v, &dst[i]);
    }
}

// -----------------------------------------------------------------------------
// Kernel 2: one block per (b, c). Stage the 34x34 input tile into LDS via the
// CDNA5 async global->LDS DMA (ASYNCcnt path), then compute the 17x17 pooled
// window from LDS and scatter it into the output. Block (b, c==0) also writes
// the new (oy, ox) location pair.
// -----------------------------------------------------------------------------
__global__ void inc_maxpool_patch_kernel(const float* __restrict__ in,
                                         const int* __restrict__ loc,
                                         float* __restrict__ out,
                                         int* __restrict__ loc_out) {
    __shared__ float tile[EXT_H * EXT_W];   // 34*34*4 = 4624 bytes

    const int bc = blockIdx.x;
    const int b = bc >> 6;          // / CC
    const int c = bc & 63;          // % CC
    const int tid = threadIdx.x;

    // oy = clip(ceil_div(loc0 - k + 1, s), 0, Ho-17). For loc0 >= 0, s=2, k=2
    // this is exactly loc0/2 (C truncation == ceil((loc0-1)/2) for loc0 >= 0).
    int l0 = loc[2 * b + 0];
    int l1 = loc[2 * b + 1];
    int oy = l0 >> 1;
    int ox = l1 >> 1;
    oy = min(max(oy, 0), HO - OUT_PH);
    ox = min(max(ox, 0), WO - OUT_PW);
    const int iy0 = oy * 2;         // top-left of the 34x34 input tile
    const int ix0 = ox * 2;         // even -> 8-byte aligned rows

    const float* src = in + (size_t)(b * CC + c) * (HH * WW);

    // ---- async DMA: 34 rows x 17 qwords (8B chunks) into LDS ---------------
    const unsigned lds_base = (unsigned)(uintptr_t)&tile[0]; // low 32b = LDS offset
    for (int t = tid; t < EXT_H * (EXT_W / 2); t += blockDim.x) {
        const int r  = t / (EXT_W / 2);
        const int c2 = t % (EXT_W / 2);
        const float* g = src + (size_t)(iy0 + r) * WW + (ix0 + 2 * c2);
        const unsigned lds_addr = lds_base + (unsigned)(t * 8);
        asm volatile("global_load_async_to_lds_b64 %0, %1, off"
                     :: "v"(lds_addr), "v"(g) : "memory");
    }
    // Drain this wave's async DMA, then sync the workgroup so every wave sees
    // the full tile in LDS.
    asm volatile("s_wait_asynccnt 0" ::: "memory");
    __syncthreads();

    // ---- 2x2/stride-2 max over the LDS tile --> 17x17 outputs --------------
    float* dst = out + (size_t)(b * CC + c) * (HO * WO);
    for (int t = tid; t < OUT_PH * OUT_PW; t += blockDim.x) {
        const int i = t / OUT_PW;
        const int j = t % OUT_PW;
        const float* r0 = &tile[(2 * i) * EXT_W + 2 * j];
        const float a0 = r0[0],      a1 = r0[1];
        const float b0 = r0[EXT_W],  b1 = r0[EXT_W + 1];
        const float m = fmaxf(fmaxf(a0, a1), fmaxf(b0, b1));
        dst[(size_t)(oy + i) * WO + (ox + j)] = m;
    }

    if (c == 0 && tid == 0) {
        loc_out[2 * b + 0] = oy;
        loc_out[2 * b + 1] = ox;
    }
}

// -----------------------------------------------------------------------------
// Host-side launcher (graph-capture safe: only kernel launches on `stream`).
// d_in order: in_tensor(f32), out_tensor(f32), patch_location(i32), then
// scalar params (ignored; shapes/params are static).
// d_out layout: new_out (25,690,112 f32) || new_loc (64 i32, raw bits).
// -----------------------------------------------------------------------------
extern "C" void kernel_launch(void* const* d_in, const int* in_sizes, int n_in,
                              void* d_out, int out_size, void* d_ws, size_t ws_size,
                              hipStream_t stream) {
    const float* in_tensor = (const float*)d_in[0];
    const float* prev_out  = (const float*)d_in[1];
    const int*   loc       = (const int*)d_in[2];

    float* out     = (float*)d_out;
    int*   loc_out = (int*)(out + (size_t)N_OUT);

    // Bulk copy: N_OUT/4 float4s; N_OUT = 25,690,112 -> 6,422,528 vec4 -> 25088 blocks.
    const int n4 = N_OUT / 4;
    const int threads = 256;
    const int blocks = (n4 + threads - 1) / threads;
    inc_maxpool_copy_kernel<<<blocks, threads, 0, stream>>>(
        (const f4*)prev_out, (f4*)out, n4);

    // Patch recompute + scatter: one block per (b, c).
    inc_maxpool_patch_kernel<<<BB * CC, 256, 0, stream>>>(
        in_tensor, loc, out, loc_out);
}